// GCN_18004502905472
// MI455X (gfx1250) — compile-verified
//
#include <hip/hip_runtime.h>

typedef float v2f __attribute__((ext_vector_type(2)));
typedef float v8f __attribute__((ext_vector_type(8)));

#define IN_C 64
#define HID_C 64
#define OUT_C 40

static __device__ __forceinline__ void atomic_add_f32(float* p, float v) {
  // native global_atomic_add_f32 (L2 atomic units), no CAS loop
  unsafeAtomicAdd(p, v);
}

// ---------------- degree / normalization ----------------
__global__ void k_deg_init(float* __restrict__ deg, int N) {
  int i = blockIdx.x * blockDim.x + threadIdx.x;
  if (i < N) deg[i] = 1.0f;  // self-loop contributes 1
}

__global__ void k_deg_accum(const int* __restrict__ dst, float* __restrict__ deg, int E) {
  int e = blockIdx.x * blockDim.x + threadIdx.x;
  if (e < E) atomic_add_f32(&deg[dst[e]], 1.0f);
}

__global__ void k_dinv(float* __restrict__ deg, int N) {
  int i = blockIdx.x * blockDim.x + threadIdx.x;
  if (i < N) deg[i] = rsqrtf(deg[i]);  // deg >= 1 always (self-loop)
}

__global__ void k_norm(const int* __restrict__ src, const int* __restrict__ dst,
                       const float* __restrict__ dinv, float* __restrict__ nrm, int E) {
  int e = blockIdx.x * blockDim.x + threadIdx.x;
  if (e < E) nrm[e] = dinv[src[e]] * dinv[dst[e]];
}

// ---------------- dense GEMM via fp32 WMMA ----------------
// T[N][NOUT] = X[N][64] @ W[64][NOUT]; block = 256 thr (8 waves), 128 rows/block.
__launch_bounds__(256)
__global__ void k_gemm_wmma(const float* __restrict__ X, const float* __restrict__ W,
                            float* __restrict__ T, int N, int NOUT, int NTILES) {
  __shared__ __align__(16) float As[128 * 68];  // stride 68 -> conflict-free column reads
  __shared__ __align__(16) float Ws[64 * 64];   // zero-padded to 64 cols
  const int tid  = threadIdx.x;
  const int row0 = blockIdx.x * 128;

  for (int i = tid; i < 64 * 64; i += 256) {
    int k = i >> 6, n = i & 63;
    Ws[i] = (n < NOUT) ? W[k * NOUT + n] : 0.0f;
  }
  for (int i = tid; i < 128 * 16; i += 256) {  // float4-granular staging of X strip
    int r = i >> 4, q = i & 15;
    int grow = row0 + r;
    float4 v = make_float4(0.f, 0.f, 0.f, 0.f);
    if (grow < N) v = *reinterpret_cast<const float4*>(X + (size_t)grow * 64 + q * 4);
    *reinterpret_cast<float4*>(&As[r * 68 + q * 4]) = v;
  }
  __syncthreads();

  const int wave = tid >> 5;
  const int lane = tid & 31;
  const int m    = lane & 15;
  const int half = lane >> 4;
  const int kb   = half << 1;                 // A/B frag: lanes 16-31 hold K+2,K+3
  const float* arow = &As[(wave * 16 + m) * 68];

  for (int nt = 0; nt < NTILES; ++nt) {
    const int n = nt * 16 + m;
    v8f acc = {};
#pragma unroll
    for (int k0 = 0; k0 < 64; k0 += 4) {
      const int k = k0 + kb;
      v2f a, b;
      a.x = arow[k];
      a.y = arow[k + 1];
      b.x = Ws[k * 64 + n];
      b.y = Ws[(k + 1) * 64 + n];
      acc = __builtin_amdgcn_wmma_f32_16x16x4_f32(false, a, false, b,
                                                  (short)0, acc, false, false);
    }
    if (n < NOUT) {
#pragma unroll
      for (int j = 0; j < 8; ++j) {           // C/D layout: M = j + 8*half
        int grow = row0 + wave * 16 + half * 8 + j;
        if (grow < N) T[(size_t)grow * NOUT + n] = acc[j];
      }
    }
  }
}

// ---------------- aggregation ----------------
__global__ void k_self_init(const float* __restrict__ t, const float* __restrict__ dinv,
                            float* __restrict__ out, int total, int C) {
  int idx = blockIdx.x * blockDim.x + threadIdx.x;
  if (idx >= total) return;
  int i = idx / C;
  float di = dinv[i];
  out[idx] = di * di * t[idx];
}

// 256 threads = 4 edges x 64 channel-lanes; wave = 32 contiguous channels of one edge
__global__ void k_scatter(const int* __restrict__ src, const int* __restrict__ dst,
                          const float* __restrict__ nrm, const float* __restrict__ t,
                          float* __restrict__ out, int E, int C) {
  int c = threadIdx.x & 63;
  int e = blockIdx.x * 4 + (threadIdx.x >> 6);
  if (e >= E || c >= C) return;
  int s = src[e], d = dst[e];
  atomic_add_f32(&out[(size_t)d * C + c], nrm[e] * t[(size_t)s * C + c]);
}

__global__ void k_bias_relu(float* __restrict__ h, const float* __restrict__ b,
                            int total, int C) {
  int idx = blockIdx.x * blockDim.x + threadIdx.x;
  if (idx >= total) return;
  int c = idx % C;
  h[idx] = fmaxf(h[idx] + b[c], 0.0f);
}

// ---------------- final bias + log_softmax (40 classes, one wave per row) ----------
__launch_bounds__(256)
__global__ void k_log_softmax40(const float* __restrict__ agg, const float* __restrict__ b,
                                float* __restrict__ out, int N) {
  int lane = threadIdx.x & 31;
  int row  = blockIdx.x * 8 + (threadIdx.x >> 5);
  if (row >= N) return;
  const float* r = agg + (size_t)row * 40;
  float v1 = r[lane] + b[lane];                                    // lanes 0..31 < 40
  float v2 = (lane < 8) ? (r[32 + lane] + b[32 + lane]) : -__builtin_inff();
  float mx = fmaxf(v1, v2);
#pragma unroll
  for (int off = 16; off >= 1; off >>= 1) mx = fmaxf(mx, __shfl_xor(mx, off, 32));
  float s = expf(v1 - mx) + ((lane < 8) ? expf(v2 - mx) : 0.0f);
#pragma unroll
  for (int off = 16; off >= 1; off >>= 1) s += __shfl_xor(s, off, 32);
  float lse = mx + logf(s);
  out[(size_t)row * 40 + lane] = v1 - lse;
  if (lane < 8) out[(size_t)row * 40 + 32 + lane] = v2 - lse;
}

// ---------------- launch ----------------
extern "C" void kernel_launch(void* const* d_in, const int* in_sizes, int n_in,
                              void* d_out, int out_size, void* d_ws, size_t ws_size,
                              hipStream_t stream) {
  const float* x  = (const float*)d_in[0];
  const int*   ei = (const int*)d_in[1];
  const float* W1 = (const float*)d_in[2];
  const float* b1 = (const float*)d_in[3];
  const float* Wh = (const float*)d_in[4];
  const float* bh = (const float*)d_in[5];
  const float* W2 = (const float*)d_in[6];
  const float* b2 = (const float*)d_in[7];
  float* out = (float*)d_out;

  const int N = in_sizes[0] / IN_C;
  const int E = in_sizes[1] / 2;
  const int* src = ei;
  const int* dst = ei + E;

  float* wsf  = (float*)d_ws;
  float* dinv = wsf;                       // N
  float* nrm  = dinv + N;                  // E
  float* t    = nrm + E;                   // N*64
  float* agg  = t + (size_t)N * 64;        // N*64

  auto cdiv = [](int a, int b) { return (a + b - 1) / b; };

  // GCN normalization: dinv = rsqrt(indeg + 1); per-edge norm
  k_deg_init<<<cdiv(N, 256), 256, 0, stream>>>(dinv, N);
  k_deg_accum<<<cdiv(E, 256), 256, 0, stream>>>(dst, dinv, E);
  k_dinv<<<cdiv(N, 256), 256, 0, stream>>>(dinv, N);
  k_norm<<<cdiv(E, 256), 256, 0, stream>>>(src, dst, dinv, nrm, E);

  // layer 1: relu(Ahat (x@W1) + b1)
  k_gemm_wmma<<<cdiv(N, 128), 256, 0, stream>>>(x, W1, t, N, HID_C, 4);
  k_self_init<<<cdiv(N * HID_C, 256), 256, 0, stream>>>(t, dinv, agg, N * HID_C, HID_C);
  k_scatter<<<cdiv(E, 4), 256, 0, stream>>>(src, dst, nrm, t, agg, E, HID_C);
  k_bias_relu<<<cdiv(N * HID_C, 256), 256, 0, stream>>>(agg, b1, N * HID_C, HID_C);

  // layer 2 (hidden): relu(Ahat (h@Wh) + bh)
  k_gemm_wmma<<<cdiv(N, 128), 256, 0, stream>>>(agg, Wh, t, N, HID_C, 4);
  k_self_init<<<cdiv(N * HID_C, 256), 256, 0, stream>>>(t, dinv, agg, N * HID_C, HID_C);
  k_scatter<<<cdiv(E, 4), 256, 0, stream>>>(src, dst, nrm, t, agg, E, HID_C);
  k_bias_relu<<<cdiv(N * HID_C, 256), 256, 0, stream>>>(agg, bh, N * HID_C, HID_C);

  // layer 3: log_softmax(Ahat (h@W2) + b2)
  k_gemm_wmma<<<cdiv(N, 128), 256, 0, stream>>>(agg, W2, t, N, OUT_C, 3);
  k_self_init<<<cdiv(N * OUT_C, 256), 256, 0, stream>>>(t, dinv, agg, N * OUT_C, OUT_C);
  k_scatter<<<cdiv(E, 4), 256, 0, stream>>>(src, dst, nrm, t, agg, E, OUT_C);
  k_log_softmax40<<<cdiv(N, 8), 256, 0, stream>>>(agg, b2, out, N);
}